// GCNGraph_49280454754826
// MI455X (gfx1250) — compile-verified
//
#include <hip/hip_runtime.h>
#include <hip/hip_bf16.h>

// ---------------------------------------------------------------------------
// GCN + ASAP pooling pipeline for MI455X (gfx1250).
// All GEMMs run on the fp32 WMMA pipe (v_wmma_f32_16x16x4_f32); each wave
// computes a 16x32 C tile (one A fragment, two B fragments, 2 WMMA/K-step),
// K unrolled by 8, batched over the 16 per-graph adjacency blocks.
// ---------------------------------------------------------------------------

typedef float v2f __attribute__((ext_vector_type(2)));
typedef float v8f __attribute__((ext_vector_type(8)));

#define GRAPHS 16
#define HDIM   128

// ---------------- fp32 WMMA GEMM: C = op(A) @ B, batched -------------------
// TRANSA==0: C[m][n] = sum_k A[m*lda+k] * B[k*ldb+n]
// TRANSA==1: C[m][n] = sum_k A[k*lda+m] * B[k*ldb+n]
// Grid: (N/32, M/16, batch); block: 32 threads (one wave).
// Requires M%16==0, N%32==0, K%8==0 (guaranteed by caller).
template <int TRANSA>
__global__ void wmma_gemm_t(const float* __restrict__ Ain,
                            const float* __restrict__ Bin,
                            float* __restrict__ Cout,
                            int K, int lda, int ldb, int ldc,
                            long long sA, long long sB, long long sC)
{
    const float* A = Ain + (long long)blockIdx.z * sA;
    const float* B = Bin + (long long)blockIdx.z * sB;
    float*       C = Cout + (long long)blockIdx.z * sC;

    const int lane = threadIdx.x;      // 0..31
    const int half = lane >> 4;        // 0 or 1 (K-pair select)
    const int l16  = lane & 15;
    const int tm = blockIdx.y * 16;
    const int tn = blockIdx.x * 32;

    // Per-lane base pointers; advance by constant strides in the K loop.
    const float* ap;
    if (TRANSA) ap = A + (long long)(2 * half) * lda + tm + l16;         // rows ka, ka+1
    else        ap = A + (long long)(tm + l16) * lda + 2 * half;         // contiguous K pair
    const float* bp = B + (long long)(2 * half) * ldb + tn + l16;        // tile0 col

    v8f acc0 = {}, acc1 = {};
    for (int k0 = 0; k0 < K; k0 += 8) {
        // Speculative prefetch of the B panel two steps ahead (dropped if OOB).
        __builtin_prefetch(bp + (long long)8 * ldb);
#pragma unroll
        for (int u = 0; u < 2; ++u) {
            v2f a, b0, b1;
            if (TRANSA) {
                a.x = ap[0];
                a.y = ap[lda];
                ap += (long long)4 * lda;
            } else {
                a = *(const v2f*)ap;   // 8B-aligned: lda even, offset 2*half
                ap += 4;
            }
            b0.x = bp[0];
            b0.y = bp[ldb];
            b1.x = bp[16];
            b1.y = bp[ldb + 16];
            bp += (long long)4 * ldb;
            acc0 = __builtin_amdgcn_wmma_f32_16x16x4_f32(
                false, a, false, b0, (short)0, acc0, false, false);
            acc1 = __builtin_amdgcn_wmma_f32_16x16x4_f32(
                false, a, false, b1, (short)0, acc1, false, false);
        }
    }
#pragma unroll
    for (int r = 0; r < 8; ++r) {
        float* cr = C + (long long)(tm + r + 8 * half) * ldc + l16;
        cr[tn]      = acc0[r];
        cr[tn + 16] = acc1[r];
    }
}

// ---------------- stage-1 block packing ------------------------------------
// bA[g][i][j] = self-looped normalized-GCN adjacency block, bM = (mask|eye).
__global__ void pack_blocks(const float* __restrict__ adj,
                            const unsigned char* __restrict__ mask,
                            float* __restrict__ bA, float* __restrict__ bM,
                            int n, int Ntot)
{
    long long id = (long long)blockIdx.x * blockDim.x + threadIdx.x;
    long long tot = (long long)GRAPHS * n * n;
    if (id >= tot) return;
    int j = (int)(id % n);
    int i = (int)((id / n) % n);
    int g = (int)(id / ((long long)n * n));
    long long gi = (long long)g * n + i;
    long long gj = (long long)g * n + j;
    unsigned char mk = mask[gi * Ntot + gj];
    float a = mk ? adj[gi * Ntot + gj] : 0.f;
    if (i == j && !mk) a = 1.f;        // eye & ~mask -> 1.0
    bA[id] = a;
    bM[id] = (mk || i == j) ? 1.f : 0.f;
}

// column sums of bA -> dinv = rsqrt(deg); column sums of bM -> mdeg
__global__ void col_stats(const float* __restrict__ bA,
                          const float* __restrict__ bM,
                          float* __restrict__ dinv, float* __restrict__ mdeg,
                          int n)
{
    int id = blockIdx.x * blockDim.x + threadIdx.x;
    if (id >= GRAPHS * n) return;
    int g = id / n, j = id - g * n;
    const float* Ab = bA + (long long)g * n * n;
    const float* Mb = bM + (long long)g * n * n;
    float s = 0.f, sm = 0.f;
    for (int i = 0; i < n; ++i) {
        s  += Ab[(long long)i * n + j];
        sm += Mb[(long long)i * n + j];
    }
    dinv[id] = (s > 0.f) ? rsqrtf(s) : 0.f;
    mdeg[id] = sm;
}

// hs[r][c] = h[r][c] * dinv[r]
__global__ void scale_rows(const float* __restrict__ h,
                           const float* __restrict__ dinv,
                           float* __restrict__ out, int count, int cols)
{
    int id = blockIdx.x * blockDim.x + threadIdx.x;
    if (id >= count) return;
    out[id] = h[id] * dinv[id / cols];
}

// out = relu(BN(dinv[r]*t + bias))
__global__ void gcn_epilogue(const float* __restrict__ t,
                             const float* __restrict__ dinv,
                             const float* __restrict__ bias,
                             const float* __restrict__ gamma,
                             const float* __restrict__ beta,
                             const float* __restrict__ mean,
                             const float* __restrict__ var,
                             float* __restrict__ out, int count, int cols)
{
    int id = blockIdx.x * blockDim.x + threadIdx.x;
    if (id >= count) return;
    int c = id % cols;
    int r = id / cols;
    float v = dinv[r] * t[id] + bias[c];
    v = (v - mean[c]) * rsqrtf(var[c] + 1e-5f) * gamma[c] + beta[c];
    out[id] = v > 0.f ? v : 0.f;
}

// xq[j][c] = max over i with bM[i][j] of x[g*n+i][c]  (one block per column)
__global__ void neighbor_max(const float* __restrict__ x,
                             const float* __restrict__ bM,
                             float* __restrict__ xq, int n)
{
    int col = blockIdx.x;              // 0..GRAPHS*n-1
    int g = col / n, j = col - g * n;
    int c = threadIdx.x;               // 0..HDIM-1
    const float* Mb = bM + (long long)g * n * n;
    const float* xb = x + (long long)g * n * HDIM;
    float mx = -1e30f;
    for (int i = 0; i < n; ++i) {
        if (Mb[(long long)i * n + j] != 0.f) {
            float v = xb[(long long)i * HDIM + c];
            mx = v > mx ? v : mx;
        }
    }
    xq[(long long)col * HDIM + c] = mx;
}

__global__ void add_bias(float* __restrict__ x, const float* __restrict__ b,
                         int count, int cols)
{
    int id = blockIdx.x * blockDim.x + threadIdx.x;
    if (id >= count) return;
    x[id] += b[id % cols];
}

// y[r] = dot(x[r, 0:128], w)
__global__ void rowdot(const float* __restrict__ x, const float* __restrict__ w,
                       float* __restrict__ y, int rows)
{
    int r = blockIdx.x * blockDim.x + threadIdx.x;
    if (r >= rows) return;
    const float* xr = x + (long long)r * HDIM;
    float s = 0.f;
    for (int c = 0; c < HDIM; ++c) s += xr[c] * w[c];
    y[r] = s;
}

// masked leaky-relu attention + column softmax; one block (128 thr) per column
__global__ void attn_softmax(const float* __restrict__ xw2,
                             const float* __restrict__ xw1q,
                             const float* __restrict__ att_b,
                             const float* __restrict__ bM,
                             float* __restrict__ S, int n)
{
    int col = blockIdx.x;
    int g = col / n, j = col - g * n;
    const float* Mb = bM + (long long)g * n * n;
    float* Sb = S + (long long)g * n * n;
    int t = threadIdx.x, nt = blockDim.x;
    float bc = att_b[0];
    float qj = xw1q[col];
    __shared__ float red[HDIM];

    float mx = -1e30f;
    for (int i = t; i < n; i += nt) {
        if (Mb[(long long)i * n + j] != 0.f) {
            float sc = xw2[g * n + i] + qj + bc;
            sc = sc >= 0.f ? sc : 0.2f * sc;
            mx = fmaxf(mx, sc);
        }
    }
    red[t] = mx; __syncthreads();
    for (int s = nt >> 1; s > 0; s >>= 1) {
        if (t < s) red[t] = fmaxf(red[t], red[t + s]);
        __syncthreads();
    }
    mx = red[0]; __syncthreads();

    float sum = 0.f;
    for (int i = t; i < n; i += nt) {
        if (Mb[(long long)i * n + j] != 0.f) {
            float sc = xw2[g * n + i] + qj + bc;
            sc = sc >= 0.f ? sc : 0.2f * sc;
            sum += __expf(sc - mx);
        }
    }
    red[t] = sum; __syncthreads();
    for (int s = nt >> 1; s > 0; s >>= 1) {
        if (t < s) red[t] += red[t + s];
        __syncthreads();
    }
    float inv = 1.f / red[0]; __syncthreads();

    for (int i = t; i < n; i += nt) {
        float v = 0.f;
        if (Mb[(long long)i * n + j] != 0.f) {
            float sc = xw2[g * n + i] + qj + bc;
            sc = sc >= 0.f ? sc : 0.2f * sc;
            v = __expf(sc - mx) * inv;
        }
        Sb[(long long)i * n + j] = v;
    }
}

// a = xn@le1+le1b ; b2 = xn@le2 ; c3 = xn@le3+le3b
__global__ void fitness_dots(const float* __restrict__ xn,
                             const float* __restrict__ le1, const float* __restrict__ le1b,
                             const float* __restrict__ le2,
                             const float* __restrict__ le3, const float* __restrict__ le3b,
                             float* __restrict__ a, float* __restrict__ b2,
                             float* __restrict__ c3, int rows)
{
    int r = blockIdx.x * blockDim.x + threadIdx.x;
    if (r >= rows) return;
    const float* xr = xn + (long long)r * HDIM;
    float s1 = 0.f, s2 = 0.f, s3 = 0.f;
    for (int c = 0; c < HDIM; ++c) {
        float v = xr[c];
        s1 += v * le1[c]; s2 += v * le2[c]; s3 += v * le3[c];
    }
    a[r]  = s1 + le1b[0];
    b2[r] = s2;
    c3[r] = s3 + le3b[0];
}

// fit[j] = sigmoid( sum_i bM[i][j]*a[i] - mdeg[j]*b2[j] + c3[j] )
__global__ void fitness_combine(const float* __restrict__ a,
                                const float* __restrict__ b2,
                                const float* __restrict__ c3,
                                const float* __restrict__ bM,
                                const float* __restrict__ mdeg,
                                float* __restrict__ fit, int n)
{
    int col = blockIdx.x;
    int g = col / n, j = col - g * n;
    const float* Mb = bM + (long long)g * n * n;
    int t = threadIdx.x, nt = blockDim.x;
    __shared__ float red[HDIM];
    float s = 0.f;
    for (int i = t; i < n; i += nt)
        s += Mb[(long long)i * n + j] * a[g * n + i];
    red[t] = s; __syncthreads();
    for (int sh = nt >> 1; sh > 0; sh >>= 1) {
        if (t < sh) red[t] += red[t + sh];
        __syncthreads();
    }
    if (t == 0) {
        float z = red[0] - mdeg[col] * b2[col] + c3[col];
        fit[col] = 1.f / (1.f + __expf(-z));
    }
}

// per-graph top-k by iterative argmax in LDS (blockDim == n, n <= 256)
__global__ void topk_select(const float* __restrict__ fit, int* __restrict__ perm,
                            int n, int k)
{
    int g = blockIdx.x;
    int t = threadIdx.x;
    __shared__ float v[256];
    __shared__ float rv[256];
    __shared__ int   ri[256];
    v[t] = fit[g * n + t];
    __syncthreads();
    for (int it = 0; it < k; ++it) {
        rv[t] = v[t]; ri[t] = t;
        __syncthreads();
        for (int s = n >> 1; s > 0; s >>= 1) {
            if (t < s) {
                if (rv[t + s] > rv[t] ||
                    (rv[t + s] == rv[t] && ri[t + s] < ri[t])) {
                    rv[t] = rv[t + s]; ri[t] = ri[t + s];
                }
            }
            __syncthreads();
        }
        if (t == 0) perm[g * k + it] = ri[0];
        if (t == ri[0]) v[t] = -1e31f;
        __syncthreads();
    }
}

// x_out[g*k+p][c] = x_new[g*n+perm][c] * fit[g*n+perm]
__global__ void gather_pool(const float* __restrict__ xn,
                            const float* __restrict__ fit,
                            const int* __restrict__ perm,
                            float* __restrict__ xo, int n, int k)
{
    int id = blockIdx.x * blockDim.x + threadIdx.x;
    int tot = GRAPHS * k * HDIM;
    if (id >= tot) return;
    int c = id % HDIM;
    int p = (id / HDIM) % k;
    int g = id / (HDIM * k);
    int src = g * n + perm[g * k + p];
    xo[id] = xn[(long long)src * HDIM + c] * fit[src];
}

// Sp[g][i][p] = S[g][i][perm[p]]
__global__ void gather_cols(const float* __restrict__ S,
                            const int* __restrict__ perm,
                            float* __restrict__ Sp, int n, int k)
{
    int id = blockIdx.x * blockDim.x + threadIdx.x;
    int tot = GRAPHS * n * k;
    if (id >= tot) return;
    int p = id % k;
    int i = (id / k) % n;
    int g = id / (k * n);
    Sp[id] = S[((long long)g * n + i) * n + perm[g * k + p]];
}

// build next-stage adjacency/mask blocks from pooled A2 / M2 scores
__global__ void finalize_pool(const float* __restrict__ A2,
                              const float* __restrict__ M2,
                              float* __restrict__ bA, float* __restrict__ bM, int k)
{
    int id = blockIdx.x * blockDim.x + threadIdx.x;
    int tot = GRAPHS * k * k;
    if (id >= tot) return;
    int q = id % k;
    int p = (id / k) % k;
    bool edge = (M2[id] > 0.f) && (p != q);
    bM[id] = (edge || p == q) ? 1.f : 0.f;
    bA[id] = (p == q) ? 1.f : (edge ? A2[id] : 0.f);
}

// per-graph mean over n nodes, then 128x10 head
__global__ void final_head(const float* __restrict__ x,
                           const float* __restrict__ lw,
                           const float* __restrict__ lb,
                           float* __restrict__ out, int n)
{
    int g = blockIdx.x;
    int t = threadIdx.x;  // 0..127
    __shared__ float xg[HDIM];
    float s = 0.f;
    for (int i = 0; i < n; ++i)
        s += x[((long long)g * n + i) * HDIM + t];
    xg[t] = s / (float)n;
    __syncthreads();
    if (t < 10) {
        float o = lb[t];
        for (int c = 0; c < HDIM; ++c) o += xg[c] * lw[c * 10 + t];
        out[g * 10 + t] = o;
    }
}

// ---------------------------------------------------------------------------
extern "C" void kernel_launch(void* const* d_in, const int* in_sizes, int n_in,
                              void* d_out, int out_size, void* d_ws, size_t ws_size,
                              hipStream_t stream)
{
    (void)in_sizes; (void)n_in; (void)out_size; (void)ws_size;

    const int N1 = 4096, n1 = 256, k1 = 128;
    const int N2 = 2048, n2 = 128, k2 = 64;
    const int N3 = 1024, n3 = 64;

    const float*         x0   = (const float*)d_in[0];
    const float*         adj  = (const float*)d_in[1];
    const unsigned char* mask = (const unsigned char*)d_in[2];
    // d_in[3] = batch (implied by block structure)
    // gcn1: 4..21, gcn2: 22..39, gcn3: 40..57 (W*3,b*3,gamma*3,beta*3,mean*3,var*3)
    const float* lin_w = (const float*)d_in[58];
    const float* lin_b = (const float*)d_in[59];
    const float* att_w = (const float*)d_in[60];
    const float* att_b = (const float*)d_in[61];
    const float* le1_w = (const float*)d_in[62];
    const float* le1_b = (const float*)d_in[63];
    const float* le2_w = (const float*)d_in[64];
    const float* le3_w = (const float*)d_in[65];
    const float* le3_b = (const float*)d_in[66];
    const float* hw    = (const float*)d_in[67];
    const float* hb    = (const float*)d_in[68];

    // ---- workspace bump allocator ----
    char* base = (char*)d_ws;
    size_t off = 0;
    auto alloc = [&](size_t bytes) -> float* {
        off = (off + 255) & ~(size_t)255;
        float* r = (float*)(base + off);
        off += bytes;
        return r;
    };
    float* bA1 = alloc((size_t)GRAPHS * n1 * n1 * 4);
    float* bM1 = alloc((size_t)GRAPHS * n1 * n1 * 4);
    float* bA2 = alloc((size_t)GRAPHS * n2 * n2 * 4);
    float* bM2 = alloc((size_t)GRAPHS * n2 * n2 * 4);
    float* bA3 = alloc((size_t)GRAPHS * n3 * n3 * 4);
    float* bM3 = alloc((size_t)GRAPHS * n3 * n3 * 4);
    float* dinv1 = alloc(N1 * 4); float* mdeg1 = alloc(N1 * 4);
    float* dinv2 = alloc(N2 * 4); float* mdeg2 = alloc(N2 * 4);
    float* dinv3 = alloc(N3 * 4); float* mdeg3 = alloc(N3 * 4);
    float* xpa   = alloc((size_t)N1 * HDIM * 4);
    float* xpb   = alloc((size_t)N1 * HDIM * 4);
    float* hbuf  = alloc((size_t)N1 * HDIM * 4);
    float* hsbuf = alloc((size_t)N1 * HDIM * 4);
    float* tbuf  = alloc((size_t)N1 * HDIM * 4);
    float* xqmax = alloc((size_t)N1 * HDIM * 4);
    float* xq    = alloc((size_t)N1 * HDIM * 4);
    float* Sbuf  = alloc((size_t)GRAPHS * n1 * n1 * 4);
    float* xnew  = alloc((size_t)N1 * HDIM * 4);
    float* Sp    = alloc((size_t)GRAPHS * n1 * k1 * 4);
    float* Tb    = alloc((size_t)GRAPHS * n1 * k1 * 4);
    float* mSp   = alloc((size_t)GRAPHS * n1 * k1 * 4);
    float* Tm    = alloc((size_t)GRAPHS * n1 * k1 * 4);
    float* A2raw = alloc((size_t)GRAPHS * k1 * k1 * 4);
    float* M2raw = alloc((size_t)GRAPHS * k1 * k1 * 4);
    float* xw2   = alloc(N1 * 4);
    float* xw1q  = alloc(N1 * 4);
    float* avec  = alloc(N1 * 4);
    float* b2vec = alloc(N1 * 4);
    float* c3vec = alloc(N1 * 4);
    float* fitv  = alloc(N1 * 4);
    int*   perm  = (int*)alloc(GRAPHS * k1 * 4);
    float* xp1   = alloc((size_t)N2 * HDIM * 4);
    float* xp2   = alloc((size_t)N3 * HDIM * 4);

    auto gemm = [&](const float* A, const float* Bm, float* C, int M, int Nc, int K,
                    int lda, int ldb, int ldc, long long sA, long long sB,
                    long long sC, int batch, int transA) {
        dim3 grid(Nc / 32, M / 16, batch);
        if (transA)
            wmma_gemm_t<1><<<grid, 32, 0, stream>>>(A, Bm, C, K, lda, ldb, ldc,
                                                    sA, sB, sC);
        else
            wmma_gemm_t<0><<<grid, 32, 0, stream>>>(A, Bm, C, K, lda, ldb, ldc,
                                                    sA, sB, sC);
    };
    auto ew = [](int count) { return dim3((count + 255) / 256); };

    // ---- GCN block: 3 layers of conv + BN + ReLU ----
    auto gcn_block = [&](const float* xstart, int din, int Nr, int n,
                         const float* bA, const float* dinv, int pbase) -> float* {
        const float* xcur = xstart;
        int dcur = din;
        float* outb = xpa;
        for (int l = 0; l < 3; ++l) {
            const float* W  = (const float*)d_in[pbase + l];
            const float* bb = (const float*)d_in[pbase + 3 + l];
            const float* ga = (const float*)d_in[pbase + 6 + l];
            const float* be = (const float*)d_in[pbase + 9 + l];
            const float* mu = (const float*)d_in[pbase + 12 + l];
            const float* va = (const float*)d_in[pbase + 15 + l];
            int cnt = Nr * HDIM;
            gemm(xcur, W, hbuf, Nr, HDIM, dcur, dcur, HDIM, HDIM, 0, 0, 0, 1, 0);
            scale_rows<<<ew(cnt), 256, 0, stream>>>(hbuf, dinv, hsbuf, cnt, HDIM);
            gemm(bA, hsbuf, tbuf, n, HDIM, n, n, HDIM, HDIM,
                 (long long)n * n, (long long)n * HDIM, (long long)n * HDIM, GRAPHS, 0);
            outb = (l & 1) ? xpb : xpa;
            gcn_epilogue<<<ew(cnt), 256, 0, stream>>>(tbuf, dinv, bb, ga, be, mu, va,
                                                      outb, cnt, HDIM);
            xcur = outb;
            dcur = HDIM;
        }
        return outb;  // xpa after 3 layers
    };

    // ---- ASAP pooling stage ----
    auto asap = [&](const float* xv, int Nr, int n, int k,
                    const float* bA, const float* bM, const float* mdeg,
                    float* bA_o, float* bM_o, float* dinv_o, float* mdeg_o,
                    float* xp_out) {
        neighbor_max<<<Nr, HDIM, 0, stream>>>(xv, bM, xqmax, n);
        gemm(xqmax, lin_w, xq, Nr, HDIM, HDIM, HDIM, HDIM, HDIM, 0, 0, 0, 1, 0);
        add_bias<<<ew(Nr * HDIM), 256, 0, stream>>>(xq, lin_b, Nr * HDIM, HDIM);
        rowdot<<<ew(Nr), 256, 0, stream>>>(xv, att_w + HDIM, xw2, Nr);
        rowdot<<<ew(Nr), 256, 0, stream>>>(xq, att_w, xw1q, Nr);
        attn_softmax<<<Nr, HDIM, 0, stream>>>(xw2, xw1q, att_b, bM, Sbuf, n);
        gemm(Sbuf, xv, xnew, n, HDIM, n, n, HDIM, HDIM,
             (long long)n * n, (long long)n * HDIM, (long long)n * HDIM, GRAPHS, 1);
        fitness_dots<<<ew(Nr), 256, 0, stream>>>(xnew, le1_w, le1_b, le2_w,
                                                 le3_w, le3_b, avec, b2vec, c3vec, Nr);
        fitness_combine<<<Nr, HDIM, 0, stream>>>(avec, b2vec, c3vec, bM, mdeg, fitv, n);
        topk_select<<<GRAPHS, n, 0, stream>>>(fitv, perm, n, k);
        gather_pool<<<ew(GRAPHS * k * HDIM), 256, 0, stream>>>(xnew, fitv, perm,
                                                               xp_out, n, k);
        gather_cols<<<ew(GRAPHS * n * k), 256, 0, stream>>>(Sbuf, perm, Sp, n, k);
        gather_cols<<<ew(GRAPHS * n * k), 256, 0, stream>>>(bM, perm, mSp, n, k);
        // A2 = Sp^T (A Sp), M2 = mSp^T (mf mSp)
        gemm(bA, Sp, Tb, n, k, n, n, k, k,
             (long long)n * n, (long long)n * k, (long long)n * k, GRAPHS, 0);
        gemm(Sp, Tb, A2raw, k, k, n, k, k, k,
             (long long)n * k, (long long)n * k, (long long)k * k, GRAPHS, 1);
        gemm(bM, mSp, Tm, n, k, n, n, k, k,
             (long long)n * n, (long long)n * k, (long long)n * k, GRAPHS, 0);
        gemm(mSp, Tm, M2raw, k, k, n, k, k, k,
             (long long)n * k, (long long)n * k, (long long)k * k, GRAPHS, 1);
        finalize_pool<<<ew(GRAPHS * k * k), 256, 0, stream>>>(A2raw, M2raw,
                                                              bA_o, bM_o, k);
        col_stats<<<ew(GRAPHS * k), 256, 0, stream>>>(bA_o, bM_o, dinv_o, mdeg_o, k);
    };

    // ---- pipeline ----
    pack_blocks<<<ew(GRAPHS * n1 * n1), 256, 0, stream>>>(adj, mask, bA1, bM1, n1, N1);
    col_stats<<<ew(GRAPHS * n1), 256, 0, stream>>>(bA1, bM1, dinv1, mdeg1, n1);

    const float* x1 = gcn_block(x0, 64, N1, n1, bA1, dinv1, 4);
    asap(x1, N1, n1, k1, bA1, bM1, mdeg1, bA2, bM2, dinv2, mdeg2, xp1);

    const float* x2 = gcn_block(xp1, HDIM, N2, n2, bA2, dinv2, 22);
    asap(x2, N2, n2, k2, bA2, bM2, mdeg2, bA3, bM3, dinv3, mdeg3, xp2);

    const float* x3 = gcn_block(xp2, HDIM, N3, n3, bA3, dinv3, 40);
    final_head<<<GRAPHS, HDIM, 0, stream>>>(x3, hw, hb, (float*)d_out, n3);
}